// SphericalConvolution_24610162606338
// MI455X (gfx1250) — compile-verified
//
#include <hip/hip_runtime.h>

// Problem constants (from the reference)
#define P_TOTAL 262144          // points reduced per (b, c_in) row
#define B_DIM   8
#define CIN     32
#define COUT    32
#define K_SH    81              // (MAX_L+1)^2 spherical-harmonic coeffs
#define ROWS    (B_DIM * CIN)   // 256 reduction rows
#define NBLK    16              // blocks per row in stage 1 (4096 blocks total)
#define T1      256             // threads per stage-1 block (8 waves on wave32)
#define M_PAD   16              // b-dimension padded to WMMA M=16

typedef float v2f __attribute__((ext_vector_type(2)));
typedef float v4f __attribute__((ext_vector_type(4)));
typedef float v8f __attribute__((ext_vector_type(8)));

// ---------------------------------------------------------------------------
// Stage 1: streaming reduction of input[b, i, :] -> partial sums.
// 256 MiB read once => pure HBM roofline (~11 us at 23.3 TB/s).
// Non-temporal b128 loads, fully unrolled, wave32 shuffle reduce, no atomics.
// ---------------------------------------------------------------------------
__global__ void __launch_bounds__(T1)
reduce_rows(const float* __restrict__ in, float* __restrict__ partial) {
    const int blk = blockIdx.x;
    const int row = blk / NBLK;             // (b * CIN + i)
    const int seg = blk % NBLK;
    const int seg_len = P_TOTAL / NBLK;     // 16384 floats per block
    const v4f* src = (const v4f*)(in + (size_t)row * P_TOTAL + (size_t)seg * seg_len);

    float acc = 0.0f;
#pragma unroll
    for (int j = 0; j < seg_len / 4 / T1; ++j) {  // 16 x b128 NT loads / thread
        v4f v = __builtin_nontemporal_load(&src[j * T1 + threadIdx.x]);
        acc += (v.x + v.y) + (v.z + v.w);
    }

    // wave32 reduce
#pragma unroll
    for (int off = 16; off > 0; off >>= 1)
        acc += __shfl_down(acc, off, 32);

    __shared__ float lds[T1 / 32];
    if ((threadIdx.x & 31) == 0) lds[threadIdx.x >> 5] = acc;
    __syncthreads();
    if (threadIdx.x == 0) {
        float s = 0.0f;
#pragma unroll
        for (int w = 0; w < T1 / 32; ++w) s += lds[w];
        partial[row * NBLK + seg] = s;
    }
}

// ---------------------------------------------------------------------------
// Stage 2: fold NBLK partials per row into a ZERO-PADDED s[16][CIN] so the
// WMMA stage can load its A fragment with no EXEC-divergent guards.
// ---------------------------------------------------------------------------
__global__ void finalize_s(const float* __restrict__ partial, float* __restrict__ sPad) {
    int idx = blockIdx.x * blockDim.x + threadIdx.x;   // 0 .. 16*CIN-1
    if (idx < M_PAD * CIN) {
        float acc = 0.0f;
        if (idx < ROWS) {
#pragma unroll
            for (int j = 0; j < NBLK; ++j) acc += partial[idx * NBLK + j];
        }
        sPad[idx] = acc;   // rows 8..15 are zero padding
    }
}

// ---------------------------------------------------------------------------
// Stage 3: out[b,o,k] = Y[k] * sum_i coef[o,i,k] * sPad[b,i]
// One wave per k-slice (grid = 81, block = 32 => EXEC all-ones for WMMA).
// All A and B fragments hoisted into registers first (loads clause + one
// wait), then 16 back-to-back V_WMMA_F32_16X16X4_F32 (f32-exact vs reference).
// ---------------------------------------------------------------------------
__global__ void __launch_bounds__(32)
sh_mix(const float* __restrict__ sPad, const float* __restrict__ coef,
       float* __restrict__ out) {
    const int k    = blockIdx.x;        // 0..80
    const int lane = threadIdx.x;       // 0..31
    const int half = lane >> 4;         // 0: K-rows {0,1} of chunk, 1: {2,3}
    const int mn   = lane & 15;         // A: row m (= b); B/D: col n (= o mod 16)

    // Y_{l,m}(0,0): sqrt((2l+1)/(4pi)) iff k == l*l + l (m == 0), else 0.
    float yk = 0.0f;
#pragma unroll
    for (int l = 0; l <= 8; ++l)
        if (k == l * l + l)
            yk = sqrtf((2.0f * l + 1.0f) * 0.07957747154594767f); // 1/(4pi)

    // ---- Hoist A fragments: lane 16*khi + m holds A[m][k0], A[m][k0+1] ----
    v2f afr[8];
#pragma unroll
    for (int kc = 0; kc < 8; ++kc) {
        const int i0 = kc * 4 + half * 2;
        afr[kc].x = sPad[mn * CIN + i0 + 0];
        afr[kc].y = sPad[mn * CIN + i0 + 1];
    }

    // ---- Hoist B fragments: lanes 0-15 rows {0,1}, lanes 16-31 rows {2,3} --
    v2f bfr[2][8];
#pragma unroll
    for (int t = 0; t < 2; ++t) {
#pragma unroll
        for (int kc = 0; kc < 8; ++kc) {
            const int i0 = kc * 4 + half * 2;
            const int o  = t * 16 + mn;
            const float* p = coef + ((size_t)o * CIN + i0) * K_SH + k;
            bfr[t][kc].x = p[0];
            bfr[t][kc].y = p[K_SH];
        }
    }

    // ---- 16 back-to-back WMMAs, then scaled store of valid b rows ---------
#pragma unroll
    for (int t = 0; t < 2; ++t) {
        v8f acc = {};
#pragma unroll
        for (int kc = 0; kc < 8; ++kc) {
            acc = __builtin_amdgcn_wmma_f32_16x16x4_f32(
                /*neg_a=*/false, afr[kc], /*neg_b=*/false, bfr[t][kc],
                /*c_mod=*/(short)0, acc, /*reuse_a=*/false, /*reuse_b=*/false);
        }
        // D element r: lanes 0-15 -> M=r (b=r, valid), lanes 16-31 -> M=r+8 (pad)
        if (half == 0) {
#pragma unroll
            for (int r = 0; r < B_DIM; ++r)
                out[((size_t)r * COUT + (t * 16 + mn)) * K_SH + k] = yk * acc[r];
        }
    }
}

// ---------------------------------------------------------------------------
extern "C" void kernel_launch(void* const* d_in, const int* in_sizes, int n_in,
                              void* d_out, int out_size, void* d_ws, size_t ws_size,
                              hipStream_t stream) {
    const float* in   = (const float*)d_in[0];   // [8, 32, 262144] f32
    const float* coef = (const float*)d_in[1];   // [32, 32, 81] f32
    float* out = (float*)d_out;                  // [8, 32, 81] f32

    float* partial = (float*)d_ws;               // ROWS * NBLK floats
    float* sPad    = partial + ROWS * NBLK;      // M_PAD * CIN floats (zero-padded)

    reduce_rows<<<ROWS * NBLK, T1, 0, stream>>>(in, partial);
    finalize_s<<<2, 256, 0, stream>>>(partial, sPad);
    sh_mix<<<K_SH, 32, 0, stream>>>(sPad, coef, out);
}